// SpGAT_25134148616268
// MI455X (gfx1250) — compile-verified
//
#include <hip/hip_runtime.h>

#define N_NODES 50000
#define NFEAT   128
#define NHID    64
#define NHEADS  8
#define C1      (NHID * NHEADS)     // 512 (layer-1 concat width)
#define NCLASS  64
#define F2      (NCLASS / NHEADS)   // 8
#define ALPHA   0.2f

typedef __attribute__((ext_vector_type(2))) float v2f;
typedef __attribute__((ext_vector_type(8))) float v8f;

// ---------------------------------------------------------------------------
// GEMM1: h1[N, 512] = x[N,128] @ Wcat1[128,512], Wcat1[:, i*64+j] = W1[i,:,j].
// One wave -> one 16x16 tile, V_WMMA_F32_16X16X4_F32, K stepped by 4.
// Each N-tile (16 cols) lies inside a single head (64 % 16 == 0).
// ---------------------------------------------------------------------------
__global__ void gat_gemm1(const float* __restrict__ x,
                          const float* __restrict__ W1,
                          float* __restrict__ h1) {
  const int lane = threadIdx.x & 31;
  const int lm   = lane & 15;          // M (A) / N (B,D) index within tile
  const int kh   = lane >> 4;          // K half: lanes 16-31 hold K+2,K+3
  const int m0   = blockIdx.y * 16;
  const int c0   = (blockIdx.x * 4 + (threadIdx.x >> 5)) * 16;  // 0..496
  const int head = c0 >> 6;
  const int j0   = c0 & 63;
  // B element [k][c0+lm] lives at W1 + head*128*64 + k*64 + (j0+lm)
  const float* wb = W1 + head * (NFEAT * NHID) + j0 + lm;
  const float* ab = x + (size_t)(m0 + lm) * NFEAT + 2 * kh;

  v8f acc = {};
  for (int k0 = 0; k0 < NFEAT; k0 += 4) {
    v2f a = *(const v2f*)(ab + k0);          // A[lm][k0+2kh], A[lm][k0+2kh+1]
    v2f b;
    b.x = wb[(k0 + 2 * kh) * NHID];          // B[k0+2kh][c0+lm]
    b.y = wb[(k0 + 2 * kh + 1) * NHID];
    acc = __builtin_amdgcn_wmma_f32_16x16x4_f32(false, a, false, b,
                                                (short)0, acc, false, false);
  }
  const int col   = c0 + lm;
  const int rbase = m0 + (kh ? 8 : 0);       // D: VGPR r -> row m0+r / m0+8+r
#pragma unroll
  for (int r = 0; r < 8; ++r)
    h1[(size_t)(rbase + r) * C1 + col] = acc[r];
}

// ---------------------------------------------------------------------------
// Per-node attention logits, layer 1: s_src[n,i] = h1[n, i*64:+64] . a1[i,:64]
//                                     s_dst[n,i] = h1[n, i*64:+64] . a1[i,64:]
// ---------------------------------------------------------------------------
__global__ void gat_s1(const float* __restrict__ h1, const float* __restrict__ a1,
                       float* __restrict__ ssrc, float* __restrict__ sdst) {
  int t = blockIdx.x * blockDim.x + threadIdx.x;
  if (t >= N_NODES * NHEADS) return;
  int n = t >> 3, i = t & 7;
  const float* hp = h1 + (size_t)n * C1 + i * NHID;
  const float* ap = a1 + i * (2 * NHID);
  float s0 = 0.f, s1 = 0.f;
#pragma unroll 8
  for (int j = 0; j < NHID; ++j) {
    float hv = hp[j];
    s0 += hv * ap[j];
    s1 += hv * ap[NHID + j];
  }
  ssrc[t] = s0;
  sdst[t] = s1;
}

// ---------------------------------------------------------------------------
// Edge aggregation, layer 1: one wave per (edge, head).
// Lane l covers columns 2l, 2l+1 of the 64-wide head slice (float2 gather,
// contiguous 256B per wave; consecutive waves walk consecutive head slices).
// ---------------------------------------------------------------------------
__global__ void gat_edge1(const int* __restrict__ src, const int* __restrict__ dst,
                          const float* __restrict__ h,
                          const float* __restrict__ ssrc, const float* __restrict__ sdst,
                          float* __restrict__ num, float* __restrict__ rowsum, int E) {
  int wid  = (blockIdx.x * blockDim.x + threadIdx.x) >> 5;
  int lane = threadIdx.x & 31;
  int edge = wid >> 3;
  int head = wid & 7;
  if (edge >= E) return;
  int s = src[edge], d = dst[edge];
  float z   = ssrc[s * NHEADS + head] + sdst[d * NHEADS + head];
  float lr  = z > 0.f ? z : ALPHA * z;
  float att = __expf(-lr);
  const float2 hv = *(const float2*)(h + (size_t)d * C1 + head * NHID + 2 * lane);
  float* np = num + (size_t)s * C1 + head * NHID + 2 * lane;
  atomicAdd(np,     att * hv.x);
  atomicAdd(np + 1, att * hv.y);
  if (lane == 0) atomicAdd(rowsum + s * NHEADS + head, att);
}

// h1act = elu(num1 / rowsum1), written in-place into num1 (layer-2 input).
__global__ void gat_fin1(float* __restrict__ num, const float* __restrict__ rowsum) {
  int i = blockIdx.x * 256 + threadIdx.x;            // i < N*512 (exact grid)
  int n = i >> 9, head = (i & 511) >> 6;
  float v = num[i] / rowsum[n * NHEADS + head];
  num[i] = v > 0.f ? v : expm1f(v);
}

// ---------------------------------------------------------------------------
// GEMM2: h2[N,64] = h1act[N,512] @ Wcat2[512,64], Wcat2[:, i*8+j] = W2[i,:,j].
// grid.y = M tiles, one block = 4 waves = all 4 N-tiles.
// ---------------------------------------------------------------------------
__global__ void gat_gemm2(const float* __restrict__ h1act,
                          const float* __restrict__ W2,
                          float* __restrict__ h2) {
  const int lane = threadIdx.x & 31;
  const int lm   = lane & 15;
  const int kh   = lane >> 4;
  const int m0   = blockIdx.y * 16;
  const int c0   = (threadIdx.x >> 5) * 16;          // 0,16,32,48
  const int col  = c0 + lm;
  const int head = col >> 3, j = col & 7;            // heads straddle tiles; per-lane base
  const float* wb = W2 + head * (C1 * F2) + j;       // B[k][col] at wb + k*F2
  const float* ab = h1act + (size_t)(m0 + lm) * C1 + 2 * kh;

  v8f acc = {};
  for (int k0 = 0; k0 < C1; k0 += 4) {
    v2f a = *(const v2f*)(ab + k0);
    v2f b;
    b.x = wb[(k0 + 2 * kh) * F2];
    b.y = wb[(k0 + 2 * kh + 1) * F2];
    acc = __builtin_amdgcn_wmma_f32_16x16x4_f32(false, a, false, b,
                                                (short)0, acc, false, false);
  }
  const int rbase = m0 + (kh ? 8 : 0);
#pragma unroll
  for (int r = 0; r < 8; ++r)
    h2[(size_t)(rbase + r) * NCLASS + col] = acc[r];
}

// Layer-2 logits: F=8 per head, a2 is [8, 16].
__global__ void gat_s2(const float* __restrict__ h2, const float* __restrict__ a2,
                       float* __restrict__ ssrc, float* __restrict__ sdst) {
  int t = blockIdx.x * blockDim.x + threadIdx.x;
  if (t >= N_NODES * NHEADS) return;
  int n = t >> 3, i = t & 7;
  const float* hp = h2 + (size_t)n * NCLASS + i * F2;
  const float* ap = a2 + i * (2 * F2);
  float s0 = 0.f, s1 = 0.f;
#pragma unroll
  for (int j = 0; j < F2; ++j) {
    float hv = hp[j];
    s0 += hv * ap[j];
    s1 += hv * ap[F2 + j];
  }
  ssrc[t] = s0;
  sdst[t] = s1;
}

// Edge aggregation, layer 2: one wave per edge, all 8 heads x 8 cols.
// Lane l -> cols 2l,2l+1 (same head since head boundary = 8); head = l/4.
__global__ void gat_edge2(const int* __restrict__ src, const int* __restrict__ dst,
                          const float* __restrict__ h2,
                          const float* __restrict__ ssrc, const float* __restrict__ sdst,
                          float* __restrict__ num, float* __restrict__ rowsum, int E) {
  int wid  = (blockIdx.x * blockDim.x + threadIdx.x) >> 5;
  int lane = threadIdx.x & 31;
  if (wid >= E) return;
  int s = src[wid], d = dst[wid];
  int head = lane >> 2;
  float z   = ssrc[s * NHEADS + head] + sdst[d * NHEADS + head];
  float lr  = z > 0.f ? z : ALPHA * z;
  float att = __expf(-lr);
  const float2 hv = *(const float2*)(h2 + (size_t)d * NCLASS + 2 * lane);
  float* np = num + (size_t)s * NCLASS + 2 * lane;
  atomicAdd(np,     att * hv.x);
  atomicAdd(np + 1, att * hv.y);
  if ((lane & 3) == 0) atomicAdd(rowsum + s * NHEADS + head, att);
}

__global__ void gat_fin2(const float* __restrict__ num, const float* __restrict__ rowsum,
                         float* __restrict__ out) {
  int i = blockIdx.x * 256 + threadIdx.x;            // i < N*64 (exact grid)
  int n = i >> 6, head = (i & 63) >> 3;
  float v = num[i] / rowsum[n * NHEADS + head];
  out[i] = v > 0.f ? v : expm1f(v);
}

// ---------------------------------------------------------------------------
extern "C" void kernel_launch(void* const* d_in, const int* in_sizes, int n_in,
                              void* d_out, int out_size, void* d_ws, size_t ws_size,
                              hipStream_t stream) {
  (void)n_in; (void)out_size; (void)ws_size;
  const float* x  = (const float*)d_in[0];
  const int*   ei = (const int*)d_in[1];
  const float* W1 = (const float*)d_in[2];
  const float* a1 = (const float*)d_in[3];
  const float* W2 = (const float*)d_in[4];
  const float* a2 = (const float*)d_in[5];
  float* out = (float*)d_out;

  const int E = in_sizes[1] / 2;
  const int* src = ei;
  const int* dst = ei + E;

  // Workspace carve-up (floats): ~238 MB total.
  float* p    = (float*)d_ws;
  float* h1   = p; p += (size_t)N_NODES * C1;
  float* s1s  = p; p += N_NODES * NHEADS;
  float* s1d  = p; p += N_NODES * NHEADS;
  float* rs1  = p; p += N_NODES * NHEADS;
  float* num1 = p; p += (size_t)N_NODES * C1;      // becomes h1act after fin1
  float* h2   = p; p += (size_t)N_NODES * NCLASS;
  float* s2s  = p; p += N_NODES * NHEADS;
  float* s2d  = p; p += N_NODES * NHEADS;
  float* rs2  = p; p += N_NODES * NHEADS;
  float* num2 = p; p += (size_t)N_NODES * NCLASS;

  hipMemsetAsync(num1, 0, sizeof(float) * (size_t)N_NODES * C1, stream);
  hipMemsetAsync(rs1,  0, sizeof(float) * N_NODES * NHEADS, stream);
  hipMemsetAsync(num2, 0, sizeof(float) * (size_t)N_NODES * NCLASS, stream);
  hipMemsetAsync(rs2,  0, sizeof(float) * N_NODES * NHEADS, stream);

  // Layer 1
  dim3 g1(C1 / 64, N_NODES / 16);                  // 8 x 3125, 4 waves/block
  gat_gemm1<<<g1, 128, 0, stream>>>(x, W1, h1);
  gat_s1<<<(N_NODES * NHEADS + 255) / 256, 256, 0, stream>>>(h1, a1, s1s, s1d);
  {
    long waves = (long)E * NHEADS;                 // wave per (edge, head)
    int blocks = (int)((waves + 7) / 8);
    gat_edge1<<<blocks, 256, 0, stream>>>(src, dst, h1, s1s, s1d, num1, rs1, E);
  }
  gat_fin1<<<(N_NODES * C1) / 256, 256, 0, stream>>>(num1, rs1);

  // Layer 2 (input = num1 in-place activated)
  dim3 g2(1, N_NODES / 16);                        // 4 waves cover all 64 cols
  gat_gemm2<<<g2, 128, 0, stream>>>(num1, W2, h2);
  gat_s2<<<(N_NODES * NHEADS + 255) / 256, 256, 0, stream>>>(h2, a2, s2s, s2d);
  {
    int blocks = (E + 7) / 8;                      // wave per edge
    gat_edge2<<<blocks, 256, 0, stream>>>(src, dst, h2, s2s, s2d, num2, rs2, E);
  }
  gat_fin2<<<(N_NODES * NCLASS) / 256, 256, 0, stream>>>(num2, rs2, out);
}